// DifferentiableMVOLayer_27650999452180
// MI455X (gfx1250) — compile-verified
//
#include <hip/hip_runtime.h>
#include <hip/hip_bf16.h>
#include <math.h>

typedef __attribute__((ext_vector_type(2))) float v2f;
typedef __attribute__((ext_vector_type(8))) float v8f;

#define N_ASSETS 512
#define BATCH    128
#define LAMBDA   10.0f
#define FISTA_ITERS 300
#define POWER_ITERS 16
#define BISECT_ITERS 32

// ---------------------------------------------------------------------------
// Kernel 1: Q[b] = U[b]^T U[b]  (batched SYRK) via V_WMMA_F32_16X16X4_F32.
// One wave per 64(j) x 64(k) tile of Q: 4 A-frags x 4 B-frags -> 16 f32
// accumulators, so each loaded fragment feeds 4 WMMAs (1.0 loads/WMMA).
// A-frag layout (16x4 f32): lanes 0-15 M=0-15; VGPR0 K={0|2}, VGPR1 K={1|3}
// split across lane halves. B-frag (4x16) mirrors it; C/D 16x16 = 8 VGPRs.
// ---------------------------------------------------------------------------
__global__ void __launch_bounds__(32)
syrk_wmma_f32(const float* __restrict__ U, float* __restrict__ Q) {
    const int bz = blockIdx.z;                 // batch
    const int j0 = blockIdx.x * 64;            // Q row tile (M group)
    const int k0 = blockIdx.y * 64;            // Q col tile (N group)
    const int l  = threadIdx.x;                // lane 0..31
    const int m  = l & 15;
    const int hf = l >> 4;                     // lane-half selects K pair

    const float* __restrict__ Ub = U + (size_t)bz * N_ASSETS * N_ASSETS;

    v8f acc[4][4];
#pragma unroll
    for (int jf = 0; jf < 4; ++jf)
#pragma unroll
        for (int kf = 0; kf < 4; ++kf) acc[jf][kf] = (v8f){};

    for (int kk = 0; kk < N_ASSETS; kk += 4) {
        const int r0 = kk + 2 * hf;            // contraction rows for this half
        const float* __restrict__ row0 = Ub + (size_t)r0 * N_ASSETS;
        const float* __restrict__ row1 = row0 + N_ASSETS;

        v2f a[4], b[4];
#pragma unroll
        for (int jf = 0; jf < 4; ++jf) {
            a[jf].x = row0[j0 + 16 * jf + m];
            a[jf].y = row1[j0 + 16 * jf + m];
        }
#pragma unroll
        for (int kf = 0; kf < 4; ++kf) {
            b[kf].x = row0[k0 + 16 * kf + m];
            b[kf].y = row1[k0 + 16 * kf + m];
        }
#pragma unroll
        for (int jf = 0; jf < 4; ++jf)
#pragma unroll
            for (int kf = 0; kf < 4; ++kf)
                acc[jf][kf] = __builtin_amdgcn_wmma_f32_16x16x4_f32(
                    false, a[jf], false, b[kf], (short)0, acc[jf][kf], false, false);
    }

    float* __restrict__ Qb = Q + (size_t)bz * N_ASSETS * N_ASSETS;
#pragma unroll
    for (int jf = 0; jf < 4; ++jf)
#pragma unroll
        for (int r = 0; r < 8; ++r) {
            const int row = j0 + 16 * jf + r + 8 * hf;   // C/D: VGPR r -> M=r(+8)
            float* qr = Qb + (size_t)row * N_ASSETS + k0 + m;
            qr[0]  = acc[jf][0][r];
            qr[16] = acc[jf][1][r];
            qr[32] = acc[jf][2][r];
            qr[48] = acc[jf][3][r];
        }
}

// ---------------------------------------------------------------------------
// Block reduction helpers (256 threads = 8 wave32s).
// ---------------------------------------------------------------------------
__device__ __forceinline__ float blockSum(float val, volatile float* red) {
    const int lane = threadIdx.x & 31;
    const int wv   = threadIdx.x >> 5;
#pragma unroll
    for (int off = 16; off; off >>= 1) val += __shfl_xor(val, off, 32);
    __syncthreads();                  // protect red[] from a previous call
    if (lane == 0) red[wv] = val;
    __syncthreads();
    float s = red[0];
#pragma unroll
    for (int w = 1; w < 8; ++w) s += red[w];
    return s;                          // identical value in every thread
}

__device__ __forceinline__ float blockMax(float val, volatile float* red) {
    const int lane = threadIdx.x & 31;
    const int wv   = threadIdx.x >> 5;
#pragma unroll
    for (int off = 16; off; off >>= 1) val = fmaxf(val, __shfl_xor(val, off, 32));
    __syncthreads();
    if (lane == 0) red[wv] = val;
    __syncthreads();
    float s = red[0];
#pragma unroll
    for (int w = 1; w < 8; ++w) s = fmaxf(s, red[w]);
    return s;
}

// Coalesced row-dot: wave handles one row; lanes read float4 (b128) chunks.
__device__ __forceinline__ float rowDot4(const float* __restrict__ row,
                                         const float4* __restrict__ x4,
                                         int lane) {
    const float4* __restrict__ row4 = reinterpret_cast<const float4*>(row);
    float acc = 0.f;
#pragma unroll
    for (int u = 0; u < 4; ++u) {
        const float4 q = row4[lane + 32 * u];     // global_load_b128
        const float4 v = x4[lane + 32 * u];       // ds_load_b128
        acc += q.x * v.x + q.y * v.y + q.z * v.z + q.w * v.w;
    }
#pragma unroll
    for (int off = 16; off; off >>= 1) acc += __shfl_xor(acc, off, 32);
    return acc;                                    // valid in lane 0
}

// ---------------------------------------------------------------------------
// Kernel 2: power iteration -> L[b] = LAMBDA * (v'Qv) + 1e-6.
// One workgroup per batch item. Rows striped over waves, cols over lanes.
// ---------------------------------------------------------------------------
__global__ void __launch_bounds__(256)
power_iter(const float* __restrict__ Q, float* __restrict__ Lout) {
    const int bz = blockIdx.x;
    const float* __restrict__ Qb = Q + (size_t)bz * N_ASSETS * N_ASSETS;

    __shared__ __align__(16) float v[N_ASSETS];
    __shared__ float qv[N_ASSETS];
    __shared__ float red[8];

    const int t = threadIdx.x, lane = t & 31, wv = t >> 5;
    const float4* v4 = reinterpret_cast<const float4*>(v);

    for (int i = t; i < N_ASSETS; i += 256)
        v[i] = 0.044194173824159216f;          // 1/sqrt(512)
    __syncthreads();

    for (int it = 0; it <= POWER_ITERS; ++it) {
        for (int i = wv; i < N_ASSETS; i += 8) {
            const float* __restrict__ row = Qb + (size_t)i * N_ASSETS;
            __builtin_prefetch(row + 8 * N_ASSETS, 0, 0);   // global_prefetch_b8
            const float acc = rowDot4(row, v4, lane);
            if (lane == 0) qv[i] = acc;
        }
        __syncthreads();
        if (it == POWER_ITERS) break;          // final matvec feeds Rayleigh

        float s = 0.f;
        for (int i = t; i < N_ASSETS; i += 256) s += qv[i] * qv[i];
        s = blockSum(s, red);
        const float inv = 1.0f / (sqrtf(s) + 1e-12f);
        for (int i = t; i < N_ASSETS; i += 256) v[i] = qv[i] * inv;
        __syncthreads();
    }

    float s = 0.f;
    for (int i = t; i < N_ASSETS; i += 256) s += v[i] * qv[i];
    s = blockSum(s, red);
    if (t == 0) Lout[bz] = LAMBDA * s + 1e-6f;
}

// ---------------------------------------------------------------------------
// Kernel 3: FISTA (300 iters) with simplex projection by bisection on theta.
// theta* in [max(v)-1, max(v)-1/n]; f(theta)=sum(max(v-theta,0)) monotone.
// Per-thread state: two coordinates (t, t+256). y lives in LDS for the matvec.
// Q (134 MB) is L2-resident, streamed once per iteration per batch block.
// ---------------------------------------------------------------------------
__global__ void __launch_bounds__(256)
fista_solve(const float* __restrict__ Q, const float* __restrict__ mu,
            const float* __restrict__ Lc, float* __restrict__ out) {
    const int bz = blockIdx.x;
    const float* __restrict__ Qb  = Q  + (size_t)bz * N_ASSETS * N_ASSETS;
    const float* __restrict__ mub = mu + (size_t)bz * N_ASSETS;

    __shared__ __align__(16) float y[N_ASSETS];
    __shared__ float g[N_ASSETS];
    __shared__ float red[8];

    const int t = threadIdx.x, lane = t & 31, wv = t >> 5;
    const int i0 = t, i1 = t + 256;
    const float4* y4 = reinterpret_cast<const float4*>(y);

    const float invL = 1.0f / Lc[bz];
    const float mu0 = mub[i0], mu1 = mub[i1];

    float w0 = 1.0f / N_ASSETS, w1 = 1.0f / N_ASSETS;
    float tk = 1.0f;
    y[i0] = w0; y[i1] = w1;
    __syncthreads();

    for (int it = 0; it < FISTA_ITERS; ++it) {
        // ---- g = Q y (LAMBDA folded in below); Q streams from L2 ----
        for (int i = wv; i < N_ASSETS; i += 8) {
            const float* __restrict__ row = Qb + (size_t)i * N_ASSETS;
            __builtin_prefetch(row + 8 * N_ASSETS, 0, 0);
            const float acc = rowDot4(row, y4, lane);
            if (lane == 0) g[i] = acc;
        }
        __syncthreads();

        const float v0 = y[i0] - (LAMBDA * g[i0] - mu0) * invL;
        const float v1 = y[i1] - (LAMBDA * g[i1] - mu1) * invL;

        // ---- projection onto the simplex via bisection on theta ----
        float lo = blockMax(fmaxf(v0, v1), red);
        float hi = lo - 1.0f / N_ASSETS;
        lo -= 1.0f;
        for (int bi = 0; bi < BISECT_ITERS; ++bi) {
            const float th = 0.5f * (lo + hi);
            float s = fmaxf(v0 - th, 0.f) + fmaxf(v1 - th, 0.f);
            s = blockSum(s, red);               // identical in all threads
            if (s > 1.0f) lo = th; else hi = th;
        }
        const float th  = 0.5f * (lo + hi);
        const float wn0 = fmaxf(v0 - th, 0.f);
        const float wn1 = fmaxf(v1 - th, 0.f);

        const float tn  = 0.5f * (1.0f + sqrtf(1.0f + 4.0f * tk * tk));
        const float mom = (tk - 1.0f) / tn;

        __syncthreads();                        // everyone done reading y
        y[i0] = wn0 + mom * (wn0 - w0);
        y[i1] = wn1 + mom * (wn1 - w1);
        w0 = wn0; w1 = wn1; tk = tn;
        __syncthreads();
    }

    // final clip (already >= 0) + renormalize
    float s = blockSum(w0 + w1, red);
    const float inv = 1.0f / (s + 1e-12f);
    out[(size_t)bz * N_ASSETS + i0] = w0 * inv;
    out[(size_t)bz * N_ASSETS + i1] = w1 * inv;
}

// ---------------------------------------------------------------------------
// Launch: inputs are mu (128*512 f32) then U (128*512*512 f32); output is
// w (128*512 f32). Workspace: Q (134.2 MB f32, fits in 192 MB L2) + L (128 f32).
// ---------------------------------------------------------------------------
extern "C" void kernel_launch(void* const* d_in, const int* in_sizes, int n_in,
                              void* d_out, int out_size, void* d_ws, size_t ws_size,
                              hipStream_t stream) {
    const float* mu = (const float*)d_in[0];
    const float* U  = (const float*)d_in[1];
    float* out = (float*)d_out;

    float* Q = (float*)d_ws;
    float* L = Q + (size_t)BATCH * N_ASSETS * N_ASSETS;

    dim3 g1(N_ASSETS / 64, N_ASSETS / 64, BATCH);   // (8, 8, 128) waves
    syrk_wmma_f32<<<g1, 32, 0, stream>>>(U, Q);
    power_iter<<<BATCH, 256, 0, stream>>>(Q, L);
    fista_solve<<<BATCH, 256, 0, stream>>>(Q, mu, L, out);
}